// STFT_13013750907101
// MI455X (gfx1250) — compile-verified
//
#include <hip/hip_runtime.h>
#include <cstdint>
#include <cmath>

// ---------------- problem constants (from reference) ----------------
#define FILT   1024          // FILTER_LENGTH
#define HOP    256
#define PADW   512           // FILTER_LENGTH/2
#define CUT    513           // FILTER_LENGTH/2 + 1
#define C2     1026          // 2*CUT channels
#define NBATCH 8
#define TLEN   2097152       // samples per batch
#define NF     8193          // frames
#define LOUT   2098176       // (NF-1)*HOP + FILT
#define NG     8196          // LOUT / HOP  (output 256-blocks)
#define NTOT1  (NBATCH * NF) // 65544 : N of forward GEMM
#define NTOT3  (NBATCH * NG) // 65568 : N of inverse GEMM
#define CPAD   1040          // channels padded to multiple of 16
#define KP3    (4 * CPAD)    // 4160 : padded K of inverse GEMM
#define SCALE_F 4.0f
#define EPS_F   1e-9f

#define LDS_STRIDE 34        // 32 payload + 2 pad DWORDs: 8B-aligned v2f, conflict-free

typedef __attribute__((ext_vector_type(2))) float        v2f;
typedef __attribute__((ext_vector_type(8))) float        v8f;
typedef __attribute__((ext_vector_type(4))) unsigned int v4ui;
typedef __attribute__((ext_vector_type(8))) int          v8si;
typedef __attribute__((ext_vector_type(4))) int          v4si;

// D = A(16x4) * B(4x16) + C, fp32 WMMA, K-step 4
__device__ __forceinline__ v8f wmma_f32(v2f a, v2f b, v8f c) {
    return __builtin_amdgcn_wmma_f32_16x16x4_f32(
        false, a, false, b, (short)0, c, false, false);
}

// Generic->LDS byte offset (LDS aperture: addr[31:0] is the LDS address)
__device__ __forceinline__ unsigned lds_off(const void* p) {
    return (unsigned)(uintptr_t)p;
}

// ---------------------------------------------------------------------
// TDM: async 2D fp32 tile load, tile = tile_y rows x 32 elems, row stride
// row_stride_elems; LDS padding 2 DWORDs per 32 -> LDS row stride 34.
// D# packing per CDNA5 ISA ch.10 (group0 128b + group1 256b, 2D tensor).
// amdgpu-toolchain builtin form: (v4u, v8i, v4i, v4i, v8i, cpol).
// ---------------------------------------------------------------------
__device__ __forceinline__ void tdm_load_2d_f32(
    unsigned lds_byte, const float* gptr,
    unsigned tile_y, unsigned long long row_stride_elems)
{
    const unsigned long long ga = (unsigned long long)(uintptr_t)gptr;
    v4ui g0;
    g0.x = 1u;                                   // count=1 (valid), load, user mode
    g0.y = lds_byte;                             // lds_addr
    g0.z = (unsigned)ga;                         // global_addr[31:0]
    g0.w = (unsigned)((ga >> 32) & 0x01FFFFFFull) | (2u << 30); // addr[56:32] | type=2

    const unsigned td0 = 32u, tile_x = 32u;
    v8si g1;
    // data_size=4B(2), pad_enable, pad_interval=32 dwords(code 4), pad_amount=2 dwords(code 1)
    g1[0] = (int)((2u << 16) | (1u << 20) | (4u << 22) | (1u << 25));
    g1[1] = (int)((td0 & 0xFFFFu) << 16);                        // tensor_dim0 lo16
    g1[2] = (int)((td0 >> 16) | ((tile_y & 0xFFFFu) << 16));     // td0 hi | tensor_dim1 lo16
    g1[3] = (int)((tile_y >> 16) | (tile_x << 16));              // td1 hi | tile_dim0
    g1[4] = (int)(tile_y & 0xFFFFu);                             // tile_dim1 (tile_dim2=0)
    g1[5] = (int)(unsigned)(row_stride_elems & 0xFFFFFFFFull);   // tensor_dim0_stride lo32
    g1[6] = (int)((row_stride_elems >> 32) & 0xFFFFull);         // stride hi16
    g1[7] = 0;
    v4si z4 = {};
    v8si z8 = {};
    __builtin_amdgcn_tensor_load_to_lds(g0, g1, z4, z4, z8, 0);
}

// =====================================================================
// Kernel 0: rearranged inverse basis  Ainv[r][kk] = inv[c, 256j + r]
//   (kk = j*CPAD + c; c >= C2 zero-padded)  ->  plain 2D tensor for TDM
// =====================================================================
__global__ __launch_bounds__(256) void build_ainv(
    const float* __restrict__ inv, float* __restrict__ ainv)
{
    int idx = blockIdx.x * 256 + threadIdx.x;      // over 256*KP3
    if (idx >= 256 * KP3) return;
    const int r  = idx / KP3;
    const int kk = idx - r * KP3;
    const int j  = kk / CPAD;
    const int c  = kk - j * CPAD;
    ainv[idx] = (c < C2) ? inv[c * FILT + j * HOP + r] : 0.0f;
}

// =====================================================================
// Kernel 1: forward STFT as GEMM (M=C2, N=NBATCH*NF, K=FILT)
//   ft[b,c,f] = sum_k fwd[c,k] * wav[b, reflect(f*HOP + k - PADW)]
//   TDM-staged A (regular tiles) and B (interior tiles), double-buffered.
// =====================================================================
__global__ __launch_bounds__(256) void stft_fwd_gemm(
    const float* __restrict__ wav,   // [8, TLEN]
    const float* __restrict__ fwd,   // [C2, FILT]
    float* __restrict__ ft)          // [8, C2, NF]
{
    __shared__ float AsmB[2][64  * LDS_STRIDE];
    __shared__ float BsmB[2][128 * LDS_STRIDE];

    const int tid  = threadIdx.x;
    const int lane = tid & 31;
    const int wave = tid >> 5;
    const int lo   = lane & 15;
    const int hi   = lane >> 4;
    const int wm   = wave & 1;      // wave grid 2(M) x 4(N)
    const int wn   = wave >> 1;

    const int mbase = blockIdx.y * 64;
    const int nbase = blockIdx.x * 128;

    const int arow = tid >> 2;          // 0..63
    const int acol = (tid & 3) * 8;
    const int bcol = tid & 127;
    const int bk   = (tid >> 7) * 16;

    const int nB = nbase + bcol;
    int bb = 0, ff = 0;
    if (nB < NTOT1) { bb = nB / NF; ff = nB - bb * NF; }

    // block-uniform TDM eligibility
    const bool aFull = (mbase + 64 <= C2);
    const int  b0    = nbase / NF;
    const int  f0    = nbase - b0 * NF;
    const bool bTdm  = (((nbase + 127) / NF) == b0) && (f0 >= 2) && (f0 + 127 <= 8190);

    auto stage = [&](int buf, int ks) {
        float* As = AsmB[buf];
        float* Bs = BsmB[buf];
        if (aFull) {
            if (wave == 0)
                tdm_load_2d_f32(lds_off(As), fwd + (size_t)mbase * FILT + ks, 64, FILT);
        } else {
            const int c = mbase + arow;
            #pragma unroll
            for (int i = 0; i < 8; ++i) {
                float v = 0.0f;
                if (c < C2) v = fwd[c * FILT + ks + acol + i];
                As[arow * LDS_STRIDE + acol + i] = v;
            }
        }
        if (bTdm) {
            if (wave == 0)
                tdm_load_2d_f32(lds_off(Bs),
                                wav + (size_t)b0 * TLEN + f0 * HOP + ks - PADW,
                                128, HOP);
        } else {
            #pragma unroll
            for (int i = 0; i < 16; ++i) {
                float v = 0.0f;
                if (nB < NTOT1) {
                    int idx = ff * HOP + ks + bk + i - PADW;
                    if (idx < 0)     idx = -idx;
                    if (idx >= TLEN) idx = 2 * TLEN - 2 - idx;
                    v = wav[(size_t)bb * TLEN + idx];
                }
                Bs[bcol * LDS_STRIDE + bk + i] = v;
            }
        }
    };

    v8f acc00 = {}, acc01 = {}, acc10 = {}, acc11 = {};

    stage(0, 0);
    if (wave == 0) __builtin_amdgcn_s_wait_tensorcnt((short)0);
    __syncthreads();

    int buf = 0;
    for (int ks = 0; ks < FILT; ks += 32) {
        if (ks + 32 < FILT) stage(buf ^ 1, ks + 32);   // DMA next slice under compute
        const float* As = AsmB[buf];
        const float* Bs = BsmB[buf];
        #pragma unroll
        for (int kb = 0; kb < 8; ++kb) {
            const int k0 = kb * 4 + 2 * hi;
            v2f a0 = *(const v2f*)&As[(wm * 32 + lo)      * LDS_STRIDE + k0];
            v2f a1 = *(const v2f*)&As[(wm * 32 + 16 + lo) * LDS_STRIDE + k0];
            v2f b0v = *(const v2f*)&Bs[(wn * 32 + lo)      * LDS_STRIDE + k0];
            v2f b1v = *(const v2f*)&Bs[(wn * 32 + 16 + lo) * LDS_STRIDE + k0];
            acc00 = wmma_f32(a0, b0v, acc00);
            acc01 = wmma_f32(a0, b1v, acc01);
            acc10 = wmma_f32(a1, b0v, acc10);
            acc11 = wmma_f32(a1, b1v, acc11);
        }
        if (wave == 0) __builtin_amdgcn_s_wait_tensorcnt((short)0);
        __syncthreads();
        buf ^= 1;
    }

    // ---- epilogue: scatter to ft[b,c,f] ----
    #pragma unroll
    for (int mi = 0; mi < 2; ++mi) {
        #pragma unroll
        for (int ni = 0; ni < 2; ++ni) {
            v8f acc = (mi == 0) ? (ni == 0 ? acc00 : acc01)
                                : (ni == 0 ? acc10 : acc11);
            const int ncol = nbase + wn * 32 + ni * 16 + lo;
            if (ncol >= NTOT1) continue;
            const int ob = ncol / NF;
            const int of = ncol - ob * NF;
            #pragma unroll
            for (int v = 0; v < 8; ++v) {
                const int c = mbase + wm * 32 + mi * 16 + v + 8 * hi;
                if (c < C2)
                    ft[((size_t)ob * C2 + c) * NF + of] = acc[v];
            }
        }
    }
}

// =====================================================================
// Kernel 2: mag/phase round-trip (elementwise, in place on ft)
// =====================================================================
__global__ __launch_bounds__(256) void stft_magphase(float* __restrict__ ft)
{
    const size_t total = (size_t)NBATCH * CUT * NF;
    size_t idx = (size_t)blockIdx.x * blockDim.x + threadIdx.x;
    if (idx >= total) return;
    const size_t per_b = (size_t)CUT * NF;
    const size_t b   = idx / per_b;
    const size_t rem = idx - b * per_b;
    const size_t c   = rem / NF;
    const size_t f   = rem - c * NF;

    float* pr = &ft[(b * C2 + c)       * NF + f];
    float* pi = &ft[(b * C2 + c + CUT) * NF + f];
    const float r  = *pr;
    const float im = *pi;
    const float mag = sqrtf(r * r + im * im);
    const float ph  = atan2f(im, r);
    *pr = mag * cosf(ph);
    *pi = mag * sinf(ph);
}

// =====================================================================
// Kernel 3: inverse transform + overlap-add + normalize as GEMM.
//   y[b,256g+r] = sum_{j,c} conc[b,c,g-j] * Ainv[r, j*CPAD+c]
//   M=256 (r), N=NBATCH*NG, K=KP3. A via TDM (always regular),
//   B via VALU gather (f=g-j window, channel padding). Double-buffered.
// =====================================================================
__global__ __launch_bounds__(256) void stft_inv_gemm(
    const float* __restrict__ conc,  // [8, C2, NF]
    const float* __restrict__ ainv,  // [256, KP3]
    const float* __restrict__ sqw,   // [FILT]
    float* __restrict__ out)         // [8, TLEN]
{
    __shared__ float AsmB[2][64  * LDS_STRIDE];
    __shared__ float BsmB[2][128 * LDS_STRIDE];

    const int tid  = threadIdx.x;
    const int lane = tid & 31;
    const int wave = tid >> 5;
    const int lo   = lane & 15;
    const int hi   = lane >> 4;
    const int wm   = wave & 1;
    const int wn   = wave >> 1;

    const int mbase = blockIdx.y * 64;     // r base (0..192)
    const int nbase = blockIdx.x * 128;

    const int bcol = tid & 127;
    const int bk   = (tid >> 7) * 16;

    const int nB = nbase + bcol;
    int bb = 0, gg = 0;
    if (nB < NTOT3) { bb = nB / NG; gg = nB - bb * NG; }

    auto stage = [&](int buf, int ks) {
        float* As = AsmB[buf];
        float* Bs = BsmB[buf];
        if (wave == 0)     // A tile: plain 2D slab of rearranged basis
            tdm_load_2d_f32(lds_off(As), ainv + (size_t)mbase * KP3 + ks, 64, KP3);
        // B tile: gathered spectrum  B[kk][(b,g)] = conc[b,c,g-j]
        #pragma unroll
        for (int i = 0; i < 16; ++i) {
            const int kk = ks + bk + i;
            const int j  = kk / CPAD;
            const int c  = kk - j * CPAD;
            const int f  = gg - j;
            float v = 0.0f;
            if (nB < NTOT3 && c < C2 && f >= 0 && f < NF)
                v = conc[((size_t)bb * C2 + c) * NF + f];
            Bs[bcol * LDS_STRIDE + bk + i] = v;
        }
    };

    v8f acc00 = {}, acc01 = {}, acc10 = {}, acc11 = {};

    stage(0, 0);
    if (wave == 0) __builtin_amdgcn_s_wait_tensorcnt((short)0);
    __syncthreads();

    int buf = 0;
    for (int ks = 0; ks < KP3; ks += 32) {
        if (ks + 32 < KP3) stage(buf ^ 1, ks + 32);
        const float* As = AsmB[buf];
        const float* Bs = BsmB[buf];
        #pragma unroll
        for (int kb = 0; kb < 8; ++kb) {
            const int k0 = kb * 4 + 2 * hi;
            v2f a0 = *(const v2f*)&As[(wm * 32 + lo)      * LDS_STRIDE + k0];
            v2f a1 = *(const v2f*)&As[(wm * 32 + 16 + lo) * LDS_STRIDE + k0];
            v2f b0v = *(const v2f*)&Bs[(wn * 32 + lo)      * LDS_STRIDE + k0];
            v2f b1v = *(const v2f*)&Bs[(wn * 32 + 16 + lo) * LDS_STRIDE + k0];
            acc00 = wmma_f32(a0, b0v, acc00);
            acc01 = wmma_f32(a0, b1v, acc01);
            acc10 = wmma_f32(a1, b0v, acc10);
            acc11 = wmma_f32(a1, b1v, acc11);
        }
        if (wave == 0) __builtin_amdgcn_s_wait_tensorcnt((short)0);
        __syncthreads();
        buf ^= 1;
    }

    // ---- epilogue: window-normalize + slice [PADW, LOUT-PADW) ----
    #pragma unroll
    for (int mi = 0; mi < 2; ++mi) {
        #pragma unroll
        for (int ni = 0; ni < 2; ++ni) {
            v8f acc = (mi == 0) ? (ni == 0 ? acc00 : acc01)
                                : (ni == 0 ? acc10 : acc11);
            const int ncol = nbase + wn * 32 + ni * 16 + lo;
            if (ncol >= NTOT3) continue;
            const int ob = ncol / NG;
            const int og = ncol - ob * NG;
            #pragma unroll
            for (int v = 0; v < 8; ++v) {
                const int r = mbase + wm * 32 + mi * 16 + v + 8 * hi; // < 256
                const long t = (long)og * HOP + r;
                if (t >= PADW && t < (long)LOUT - PADW) {
                    float wf = EPS_F;
                    #pragma unroll
                    for (int j = 0; j < 4; ++j) {
                        const int f = og - j;
                        if (f >= 0 && f < NF) wf += sqw[j * HOP + r];
                    }
                    out[(size_t)ob * TLEN + (t - PADW)] = acc[v] / wf * SCALE_F;
                }
            }
        }
    }
}

// =====================================================================
extern "C" void kernel_launch(void* const* d_in, const int* in_sizes, int n_in,
                              void* d_out, int out_size, void* d_ws, size_t ws_size,
                              hipStream_t stream) {
    const float* wav = (const float*)d_in[0];   // [8, TLEN]
    const float* fwd = (const float*)d_in[1];   // [C2, FILT]
    const float* inv = (const float*)d_in[2];   // [C2, FILT]
    const float* sqw = (const float*)d_in[3];   // [FILT]
    float* out = (float*)d_out;                 // [8, TLEN]

    float* ft   = (float*)d_ws;                         // [8, C2, NF] ~269 MB
    float* ainv = ft + (size_t)NBATCH * C2 * NF;        // [256, KP3]  ~4.3 MB

    // Stage 0: rearranged inverse basis (TDM-friendly 2D tensor)
    {
        dim3 grid((256 * KP3 + 255) / 256);
        build_ainv<<<grid, 256, 0, stream>>>(inv, ainv);
    }
    // Stage 1: forward STFT GEMM
    {
        dim3 grid((NTOT1 + 127) / 128, (C2 + 63) / 64);
        stft_fwd_gemm<<<grid, 256, 0, stream>>>(wav, fwd, ft);
    }
    // Stage 2: mag/phase round trip (in place)
    {
        const size_t total = (size_t)NBATCH * CUT * NF;
        dim3 grid((unsigned)((total + 255) / 256));
        stft_magphase<<<grid, 256, 0, stream>>>(ft);
    }
    // Stage 3: inverse GEMM + overlap-add + normalize + slice
    {
        dim3 grid((NTOT3 + 127) / 128, 256 / 64);
        stft_inv_gemm<<<grid, 256, 0, stream>>>(ft, ainv, sqw, out);
    }
}